// GNN_71270687310162
// MI455X (gfx1250) — compile-verified
//
#include <hip/hip_runtime.h>

// ---------------------------------------------------------------------------
// GraphSAGE 2-layer forward for MI455X (gfx1250, wave32).
//   agg[dst] += x[src] over edges (f32 global atomics, L2-resident),
//   out = relu?( mean*W_l + self*W_r + b )  via V_WMMA_F32_16X16X4_F32.
// N_NODES=100000 (divisible by 32), IN=HID=128, OUT=64, N_EDGES=640000.
// ---------------------------------------------------------------------------

typedef float v2f __attribute__((ext_vector_type(2)));
typedef float v8f __attribute__((ext_vector_type(8)));

#define K_DIM    128    // inner dim for both layers
#define SM_PITCH 130    // LDS row pitch: even (8B-aligned b64 pairs) + conflict-free
#define ROWS     32     // nodes per block (2 row-tiles of 16; divides 100000)

__global__ void zero_f32(float* __restrict__ p, long n) {
    long i = (long)blockIdx.x * blockDim.x + threadIdx.x;
    long stride = (long)gridDim.x * blockDim.x;
    for (; i < n; i += stride) p[i] = 0.0f;
}

__global__ void degree_kernel(const long long* __restrict__ dst,
                              float* __restrict__ cnt, int n_edges) {
    int e = blockIdx.x * blockDim.x + threadIdx.x;
    if (e < n_edges) unsafeAtomicAdd(&cnt[(int)dst[e]], 1.0f);
}

// one wave32 per edge: lane handles 4 channels (float4 gather, 4 f32 atomics)
__global__ void scatter_add(const float* __restrict__ feat,
                            const long long* __restrict__ src,
                            const long long* __restrict__ dst,
                            float* __restrict__ agg, int n_edges) {
    int e = blockIdx.x * (blockDim.x >> 5) + (threadIdx.x >> 5);
    if (e >= n_edges) return;
    int lane = threadIdx.x & 31;
    int s = (int)src[e];
    int d = (int)dst[e];
    int c = lane << 2;
    const float4 v = *(const float4*)(feat + (size_t)s * K_DIM + c);
    float* p = agg + (size_t)d * K_DIM + c;
    unsafeAtomicAdd(p + 0, v.x);
    unsafeAtomicAdd(p + 1, v.y);
    unsafeAtomicAdd(p + 2, v.z);
    unsafeAtomicAdd(p + 3, v.w);
}

// Block = (N_OUT/16) waves, ROWS(=32) nodes. Wave w owns output columns
// [16w,16w+16) for both 16-row tiles; B fragments are loaded once per k-step
// and reused across the two row-tiles (4 wmma per k-step per wave).
// N_OUT is a template constant so weight/store strides fold into immediate
// ioffsets (no 64-bit VALU address chains in the hot loop or epilogue).
// D = mean(ROWSx128) @ Wl(128 x N_OUT) + self(ROWSx128) @ Wr(128 x N_OUT) + b
template <bool RELU, int N_OUT>
__global__ void sage_gemm(const float* __restrict__ self_feat,
                          const float* __restrict__ agg,
                          const float* __restrict__ cnt,
                          const float* __restrict__ Wl,
                          const float* __restrict__ Wr,
                          const float* __restrict__ bias,
                          float* __restrict__ out) {
    __shared__ float sm_mean[ROWS * SM_PITCH];
    __shared__ float sm_self[ROWS * SM_PITCH];

    const int m0   = blockIdx.x * ROWS;
    const int tid  = threadIdx.x;
    const int lane = tid & 31;
    const int wave = tid >> 5;

    // ---- stage ROWSx128 mean & self tiles into LDS (padded pitch) --------
    for (int i = tid; i < ROWS * 32; i += (N_OUT / 16) * 32) {
        int r  = i >> 5;            // local row 0..ROWS-1
        int c4 = (i & 31) << 2;     // channel group
        int node = m0 + r;          // grid exact: node < n_nodes
        const float4 xv = *(const float4*)(self_feat + (size_t)node * K_DIM + c4);
        const float4 av = *(const float4*)(agg       + (size_t)node * K_DIM + c4);
        float inv = 1.0f / fmaxf(cnt[node], 1.0f);
        float* ps = &sm_self[r * SM_PITCH + c4];
        float* pm = &sm_mean[r * SM_PITCH + c4];
        ps[0] = xv.x; ps[1] = xv.y; ps[2] = xv.z; ps[3] = xv.w;
        pm[0] = av.x * inv; pm[1] = av.y * inv; pm[2] = av.z * inv; pm[3] = av.w * inv;
    }
    __syncthreads();

    // ---- WMMA f32 16x16x4, K-loop over 128, 2 row-tiles share B ----------
    const int n0   = wave * 16;
    const int row  = lane & 15;               // A: M index; B/D: N index
    const int koff = (lane < 16) ? 0 : 2;     // K split across lane halves

    const float* smm = sm_mean + row * SM_PITCH;
    const float* sms = sm_self + row * SM_PITCH;
    const float* wlp = Wl + koff * N_OUT + n0 + row;   // base for this lane
    const float* wrp = Wr + koff * N_OUT + n0 + row;

    v8f acc0 = {0.f, 0.f, 0.f, 0.f, 0.f, 0.f, 0.f, 0.f};
    v8f acc1 = acc0;

    for (int k = 0; k < K_DIM; k += 4) {
        const int ka = k + koff;
        v2f bl = { wlp[k * N_OUT], wlp[k * N_OUT + N_OUT] };   // rows ka, ka+1
        v2f br = { wrp[k * N_OUT], wrp[k * N_OUT + N_OUT] };
        v2f am0 = { smm[ka], smm[ka + 1] };
        v2f as0 = { sms[ka], sms[ka + 1] };
        v2f am1 = { smm[16 * SM_PITCH + ka], smm[16 * SM_PITCH + ka + 1] };
        v2f as1 = { sms[16 * SM_PITCH + ka], sms[16 * SM_PITCH + ka + 1] };
        acc0 = __builtin_amdgcn_wmma_f32_16x16x4_f32(false, am0, false, bl,
                                                     (short)0, acc0, false, false);
        acc1 = __builtin_amdgcn_wmma_f32_16x16x4_f32(false, am1, false, bl,
                                                     (short)0, acc1, false, false);
        acc0 = __builtin_amdgcn_wmma_f32_16x16x4_f32(false, as0, false, br,
                                                     (short)0, acc0, false, false);
        acc1 = __builtin_amdgcn_wmma_f32_16x16x4_f32(false, as1, false, br,
                                                     (short)0, acc1, false, false);
    }

    // ---- epilogue: bias (+ReLU), coalesced stores w/ immediate offsets ----
    const float bv = bias[n0 + row];
    const int hoff = (lane < 16) ? 0 : 8;     // C/D layout: VGPR i -> M=i / M=8+i
    float* o0 = out + (size_t)(m0 + hoff) * N_OUT + n0 + row;
#pragma unroll
    for (int i = 0; i < 8; ++i) {
        float v0 = acc0[i] + bv;
        float v1 = acc1[i] + bv;
        if (RELU) { v0 = fmaxf(v0, 0.0f); v1 = fmaxf(v1, 0.0f); }
        o0[i * N_OUT] = v0;                   // tile 0, row m0+hoff+i
        o0[(16 + i) * N_OUT] = v1;            // tile 1, row m0+16+hoff+i
    }
}

extern "C" void kernel_launch(void* const* d_in, const int* in_sizes, int n_in,
                              void* d_out, int out_size, void* d_ws, size_t ws_size,
                              hipStream_t stream) {
    const float*     x   = (const float*)d_in[0];
    const long long* ei  = (const long long*)d_in[1];   // int64 edge_index [2, E]
    const float*     W1l = (const float*)d_in[2];
    const float*     W1r = (const float*)d_in[3];
    const float*     b1  = (const float*)d_in[4];
    const float*     W2l = (const float*)d_in[5];
    const float*     W2r = (const float*)d_in[6];
    const float*     b2  = (const float*)d_in[7];
    float*           out = (float*)d_out;

    const int n_nodes = in_sizes[0] / K_DIM;   // 100000 (multiple of ROWS)
    const int n_edges = in_sizes[1] / 2;       // 640000
    const long long* src = ei;
    const long long* dst = ei + n_edges;

    // workspace: agg[N*128] | cnt[N] | h[N*128]  (all f32)
    float* agg = (float*)d_ws;
    float* cnt = agg + (size_t)n_nodes * K_DIM;
    float* h   = cnt + n_nodes;

    const int ZT = 256;
    const long agg_cnt_elems = (long)n_nodes * K_DIM + n_nodes;
    int zgrid = (int)((agg_cnt_elems + ZT * 8 - 1) / (ZT * 8)); // grid-stride, 8 elem/thread
    if (zgrid < 1) zgrid = 1;

    const int gblocks = n_nodes / ROWS;        // 3125, exact

    // ---- layer 1 ----------------------------------------------------------
    zero_f32<<<zgrid, ZT, 0, stream>>>(agg, agg_cnt_elems);            // agg + cnt
    degree_kernel<<<(n_edges + 255) / 256, 256, 0, stream>>>(dst, cnt, n_edges);
    scatter_add<<<(n_edges + 7) / 8, 256, 0, stream>>>(x, src, dst, agg, n_edges);
    sage_gemm<true, 128><<<gblocks, 256, 0, stream>>>(x, agg, cnt, W1l, W1r, b1, h);

    // ---- layer 2 ----------------------------------------------------------
    zero_f32<<<zgrid, ZT, 0, stream>>>(agg, (long)n_nodes * K_DIM);
    scatter_add<<<(n_edges + 7) / 8, 256, 0, stream>>>(h, src, dst, agg, n_edges);
    sage_gemm<false, 64><<<gblocks, 128, 0, stream>>>(h, agg, cnt, W2l, W2r, b2, out);
}